// GateHadamard_77713138253951
// MI455X (gfx1250) — compile-verified
//
#include <hip/hip_runtime.h>

// Problem constants (reference: N=20 qubits, DIM=2^20, B=16 statevectors)
#define DIM      (1 << 20)
#define NROWS    16
#define CHUNK    65536        // 2^16 floats per pass-1 block (256 KiB LDS)
#define P1_THREADS 1024
#define TILE_K   16           // high-4-bit radix
#define TILE_O   128          // offsets (columns) per pass-2 block = 8 waves * 16
#define TILE_PITCH 132        // padded LDS row pitch (floats) to dodge bank conflicts
#define SEGS_PER_ROW (65536 / TILE_O)   // = 512

typedef __attribute__((ext_vector_type(2))) float v2f;
typedef __attribute__((ext_vector_type(8))) float v8f;
typedef int g4i __attribute__((vector_size(16)));   // matches async-LDS builtin param

#define AS1 __attribute__((address_space(1)))
#define AS3 __attribute__((address_space(3)))

#ifndef __has_builtin
#define __has_builtin(x) 0
#endif

// CDNA5 async global<->LDS movers (ASYNCcnt-tracked, no VGPR staging).
#if __has_builtin(__builtin_amdgcn_global_load_async_to_lds_b128) && \
    __has_builtin(__builtin_amdgcn_global_store_async_from_lds_b128)
#define HAVE_ASYNC_LDS 1
#else
#define HAVE_ASYNC_LDS 0
#endif

#if __has_builtin(__builtin_amdgcn_s_wait_asynccnt)
#define WAIT_ASYNC0() __builtin_amdgcn_s_wait_asynccnt(0)
#else
#define WAIT_ASYNC0() asm volatile("s_wait_asynccnt 0x0" ::: "memory")
#endif

// ---------------------------------------------------------------------------
// Pass 1: FWHT over the 16 low bits. One block = one contiguous 64K-float
// chunk staged through 256 KiB of LDS (CDNA5 WGP has 320 KiB). 16 butterfly
// stages with per-stage 1/sqrt(2) scaling (matches reference numerics).
// Staging uses GLOBAL_LOAD_ASYNC_TO_LDS_B128 / GLOBAL_STORE_ASYNC_FROM_LDS_B128
// when available: HBM<->LDS direct, 512 B per wave per op, drained with one
// s_wait_asynccnt before the workgroup barrier.
// ---------------------------------------------------------------------------
__global__ __launch_bounds__(P1_THREADS)
void fwht_low16(const float* __restrict__ in, float* __restrict__ out) {
    extern __shared__ float lds[];          // 65536 floats, set at launch
    const int tid = threadIdx.x;
    const long long base = (long long)blockIdx.x * CHUNK;

    // Coalesced 128-bit transfers: 64K floats / 1024 threads = 16 float4 each.
    float4* lds4 = (float4*)lds;
#if HAVE_ASYNC_LDS
    {
        AS1 g4i* g4 = (AS1 g4i*)(in + base);     // global side (const cast away)
        AS3 g4i* l4 = (AS3 g4i*)lds;             // LDS side
#pragma unroll
        for (int i = 0; i < 16; ++i)
            __builtin_amdgcn_global_load_async_to_lds_b128(
                g4 + tid + i * P1_THREADS,
                l4 + tid + i * P1_THREADS,
                /*imm offset=*/0, /*cpol=*/0);
        WAIT_ASYNC0();
    }
#else
    {
        const float4* __restrict__ in4 = (const float4*)(in + base);
#pragma unroll
        for (int i = 0; i < 16; ++i)
            lds4[tid + i * P1_THREADS] = in4[tid + i * P1_THREADS];
    }
#endif
    __syncthreads();

    const float inv = 0.70710678118654752440f;
#pragma unroll
    for (int sb = 0; sb < 16; ++sb) {
        const int s = 1 << sb;
#pragma unroll
        for (int it = 0; it < (CHUNK / 2) / P1_THREADS; ++it) {
            const int p  = tid + it * P1_THREADS;
            const int i0 = ((p & ~(s - 1)) << 1) | (p & (s - 1));
            const int i1 = i0 + s;
            const float a = lds[i0];
            const float b = lds[i1];
            lds[i0] = (a + b) * inv;
            lds[i1] = (a - b) * inv;
        }
        __syncthreads();
    }

#if HAVE_ASYNC_LDS
    {
        AS1 g4i* g4 = (AS1 g4i*)(out + base);
        AS3 g4i* l4 = (AS3 g4i*)lds;
#pragma unroll
        for (int i = 0; i < 16; ++i)
            __builtin_amdgcn_global_store_async_from_lds_b128(
                g4 + tid + i * P1_THREADS,
                l4 + tid + i * P1_THREADS,
                /*imm offset=*/0, /*cpol=*/0);
        WAIT_ASYNC0();   // drain before end (S_ENDPGM also implies wait-idle)
    }
#else
    {
        float4* out4 = (float4*)(out + base);
#pragma unroll
        for (int i = 0; i < 16; ++i)
            out4[tid + i * P1_THREADS] = lds4[tid + i * P1_THREADS];
    }
#endif
}

// ---------------------------------------------------------------------------
// Pass 2: radix-16 WHT over the 4 high bits (stride 65536 floats) done as a
// 16x16 matrix multiply on the WMMA unit:  Y = (H16 * 0.25) * X
// (the 0.25 folds the four 1/sqrt(2) stage scales into the A matrix).
// Each wave owns a 16x16 tile (16 offsets); a 256-thread block = 8 waves
// covers 128 offsets for one statevector row. In-place on d_out.
// A-matrix f32 16x4 layout (ISA 7.12.2): lanes 0-15 -> M=lane, VGPR0=K0,
// VGPR1=K1; lanes 16-31 -> same M, K2/K3. B/C/D: rows striped across lanes,
// second half-wave offset by half the rows.
// ---------------------------------------------------------------------------
__global__ __launch_bounds__(256)
void fwht_high4_wmma(float* __restrict__ data) {
    __shared__ float tile[TILE_K * TILE_PITCH];   // 8448 B

    const int tid  = threadIdx.x;
    const int row  = blockIdx.x >> 9;             // blockIdx / SEGS_PER_ROW
    const int seg  = blockIdx.x & (SEGS_PER_ROW - 1);
    const long long rowbase = (long long)row * DIM;
    const int obase = seg * TILE_O;

    // Load 16 (high-bit values) x 128 (offsets) tile; 128-float contiguous
    // runs per k -> fully coalesced.
#pragma unroll
    for (int j = 0; j < 8; ++j) {
        const int e = j * 256 + tid;
        const int k = e >> 7;          // high-4-bit value
        const int o = e & (TILE_O - 1);
        tile[k * TILE_PITCH + o] =
            data[rowbase + (long long)k * 65536 + obase + o];
    }
    __syncthreads();

    const int lane = tid & 31;         // wave32
    const int wave = tid >> 5;
    const int m    = lane & 15;        // output row this lane contributes to (A)
    const int hi   = lane >> 4;        // half-wave select
    const int col  = wave * 16 + m;    // this lane's B/D column in the tile

    v8f c = {};                        // f32 accumulator, 8 VGPRs
#pragma unroll
    for (int kb = 0; kb < 16; kb += 4) {
        const int k0 = kb + (hi << 1); // K slice per half-wave: {0,1} / {2,3}
        const int k1 = k0 + 1;
        v2f a, b;
        // A = H16 * 0.25, synthesized per-lane: H[m][k] = (-1)^popcnt(m&k)
        a.x = (__builtin_popcount(m & k0) & 1) ? -0.25f : 0.25f;
        a.y = (__builtin_popcount(m & k1) & 1) ? -0.25f : 0.25f;
        // B = data tile slice, rows k0/k1 at this lane's column
        b.x = tile[k0 * TILE_PITCH + col];
        b.y = tile[k1 * TILE_PITCH + col];
        // D(16x16,f32) += A(16x4,f32) x B(4x16,f32)
        c = __builtin_amdgcn_wmma_f32_16x16x4_f32(
                /*neg_a=*/false, a, /*neg_b=*/false, b,
                /*c_mod=*/(short)0, c, /*reuse_a=*/false, /*reuse_b=*/false);
    }

    // Scatter D back into the tile. C/D layout: VGPR v holds M=v (lanes 0-15)
    // and M=v+8 (lanes 16-31). Each wave writes only its own 16 columns, and
    // its reads above already landed in registers -> no extra barrier needed
    // before the per-wave writes.
#pragma unroll
    for (int v = 0; v < 8; ++v) {
        const int mo = v + (hi << 3);
        tile[mo * TILE_PITCH + col] = c[v];
    }
    __syncthreads();

    // Coalesced store back (in-place, disjoint element set per block).
#pragma unroll
    for (int j = 0; j < 8; ++j) {
        const int e = j * 256 + tid;
        const int k = e >> 7;
        const int o = e & (TILE_O - 1);
        data[rowbase + (long long)k * 65536 + obase + o] =
            tile[k * TILE_PITCH + o];
    }
}

// ---------------------------------------------------------------------------
// Launch: two passes, 256 MB total HBM traffic (~11 us at 23.3 TB/s) vs the
// reference's 20 passes (~2.7 GB). signs/indxs inputs are unused: the
// butterfly structure is pure bit arithmetic.
// ---------------------------------------------------------------------------
extern "C" void kernel_launch(void* const* d_in, const int* in_sizes, int n_in,
                              void* d_out, int out_size, void* d_ws, size_t ws_size,
                              hipStream_t stream) {
    (void)in_sizes; (void)n_in; (void)d_ws; (void)ws_size; (void)out_size;
    const float* x = (const float*)d_in[0];
    float* out = (float*)d_out;

    // Pass 1: bits 0..15, 256 blocks (16 rows * 16 chunks), 256 KiB dynamic LDS.
    fwht_low16<<<dim3((NROWS * DIM) / CHUNK), dim3(P1_THREADS),
                 CHUNK * sizeof(float), stream>>>(x, out);

    // Pass 2: bits 16..19 via WMMA, 16 rows * 512 segments = 8192 blocks.
    fwht_high4_wmma<<<dim3(NROWS * SEGS_PER_ROW), dim3(256), 0, stream>>>(out);
}